// Attention_26027501814371
// MI455X (gfx1250) — compile-verified
//
#include <hip/hip_runtime.h>
#include <math.h>

typedef __attribute__((ext_vector_type(2))) float v2f;
typedef __attribute__((ext_vector_type(4))) float v4f;
typedef __attribute__((ext_vector_type(8))) float v8f;

#define K_N  32
#define B_N  16384
#define DA_N 128

constexpr int ROWS_PER_TILE   = 16;
constexpr int K_SPLIT         = 2;    // waves cooperating on one tile (split over k)
constexpr int TILES_PER_BLOCK = 4;
constexpr int WAVES_PER_BLOCK = TILES_PER_BLOCK * K_SPLIT;   // 8 -> 256 threads
constexpr int LDS_STRIDE      = DA_N + 4;   // 132 floats: pad to dodge LDS bank conflicts

// ---------------- Phase 1: column-wise softmax over K ----------------
// neigh_effect [K,B] -> w [K,B] in workspace. One thread per column c.
__global__ __launch_bounds__(256) void softmax_kernel(const float* __restrict__ ne,
                                                      float* __restrict__ w) {
    int c = blockIdx.x * blockDim.x + threadIdx.x;   // 0..B-1
    float v[K_N];
    float m = -3.402823466e38f;
#pragma unroll
    for (int k = 0; k < K_N; ++k) {
        v[k] = ne[k * B_N + c];
        m = fmaxf(m, v[k]);
    }
    float s = 0.f;
#pragma unroll
    for (int k = 0; k < K_N; ++k) {
        v[k] = __expf(v[k] - m);
        s += v[k];
    }
    float inv = 1.0f / s;
#pragma unroll
    for (int k = 0; k < K_N; ++k) w[k * B_N + c] = v[k] * inv;
}

// ---------------- Phase 2: fused weighted-sum + effect + transform ----------------
// Two waves share a 16-row tile (k split 0..15 / 16..31) for 2x occupancy;
// lane l owns cols 4l..4l+3 (coalesced float4). Survivor wave runs the WMMA epilogue.
__global__ __launch_bounds__(32 * WAVES_PER_BLOCK) void fused_kernel(
    const float* __restrict__ nt,       // [K,B,DA]
    const float* __restrict__ w,        // [K,B] softmax weights
    const float* __restrict__ n_param,  // [DA]
    const float* __restrict__ ew,       // [DA]
    const float* __restrict__ eb,       // [1]
    const float* __restrict__ tw,       // [DA]
    const float* __restrict__ tb,       // [DA]
    float* __restrict__ out_effect,     // [B]
    float* __restrict__ out_y)          // [B,DA]
{
    __shared__ float s_x[TILES_PER_BLOCK][ROWS_PER_TILE * LDS_STRIDE];
    __shared__ float s_ew[DA_N];

    const int tid = threadIdx.x;
    if (tid < DA_N) s_ew[tid] = ew[tid];

    const int lane = tid & 31;
    const int wave = tid >> 5;
    const int pair = wave >> 1;         // tile index within block
    const int half = wave & 1;          // which k-half this wave owns
    const int tileBase = (blockIdx.x * TILES_PER_BLOCK + pair) * ROWS_PER_TILE;
    const int c0 = lane * 4;            // column offset of this lane's float4

    v4f acc[ROWS_PER_TILE];
#pragma unroll
    for (int r = 0; r < ROWS_PER_TILE; ++r) acc[r] = (v4f){0.f, 0.f, 0.f, 0.f};

    // ---- bandwidth phase: partial sum over this wave's k-half ----
    // nt is a 256MB single-use stream: non-temporal so it doesn't wash L2
    // (w lives in L2 and is re-read ~128x per cacheline).
    const int k0 = half * (K_N / K_SPLIT);
    for (int k = k0; k < k0 + K_N / K_SPLIT; ++k) {
        const float* ntk = nt + (size_t)k * (B_N * DA_N);
        const float* wk  = w  + (size_t)k * B_N;
#pragma unroll
        for (int r = 0; r < ROWS_PER_TILE; ++r) {
            const int row   = tileBase + r;
            const int cbase = (row * DA_N) & (B_N - 1);   // (row*128) mod 16384; +d never wraps
            const v4f t4 = __builtin_nontemporal_load(
                reinterpret_cast<const v4f*>(ntk + (size_t)row * DA_N + c0));
            const v4f w4 = *reinterpret_cast<const v4f*>(wk + cbase + c0);
            acc[r].x = fmaf(t4.x, w4.x, acc[r].x);
            acc[r].y = fmaf(t4.y, w4.y, acc[r].y);
            acc[r].z = fmaf(t4.z, w4.z, acc[r].z);
            acc[r].w = fmaf(t4.w, w4.w, acc[r].w);
        }
    }

    // ---- combine the two k-halves through LDS ----
    float* sx = s_x[pair];
    if (half == 1) {
#pragma unroll
        for (int r = 0; r < ROWS_PER_TILE; ++r)
            *reinterpret_cast<v4f*>(&sx[r * LDS_STRIDE + c0]) = acc[r];
    }
    __syncthreads();          // also covers the s_ew preload
    if (half == 1) return;    // partner wave done; survivor runs the epilogue

    // per-lane broadcast params for cols c0..c0+3
    const v4f np4 = *reinterpret_cast<const v4f*>(n_param + c0);
    const v4f tw4 = *reinterpret_cast<const v4f*>(tw + c0);
    const v4f tb4 = *reinterpret_cast<const v4f*>(tb + c0);
    const float ebv = eb[0];

    // x_ = n_param * (acc_half0 + acc_half1); restage tile into LDS (A-layout source)
#pragma unroll
    for (int r = 0; r < ROWS_PER_TILE; ++r) {
        const v4f p = *reinterpret_cast<const v4f*>(&sx[r * LDS_STRIDE + c0]);
        acc[r] = (acc[r] + p) * np4;
        *reinterpret_cast<v4f*>(&sx[r * LDS_STRIDE + c0]) = acc[r];
    }
    // same-wave LDS ops are in-order (DScnt); no further barrier needed

    // ---- dot(x_[row,:], ew) for the 16 rows via V_WMMA_F32_16X16X4_F32 ----
    // A chunk j (16x4): lane l holds row (l&15), cols {4j+2h, 4j+2h+1}, h = l>>4.
    // B chunk j (4x16): ew[4j+kk] replicated across all 16 N columns.
    const int h2   = (lane >> 4) << 1;   // 0 or 2
    const int rsel = lane & 15;
    v8f d = {};
#pragma unroll
    for (int j = 0; j < DA_N / 4; ++j) {
        const int col = 4 * j + h2;
        v2f a, bm;
        a.x  = sx[rsel * LDS_STRIDE + col];
        a.y  = sx[rsel * LDS_STRIDE + col + 1];
        bm.x = s_ew[col];
        bm.y = s_ew[col + 1];
        d = __builtin_amdgcn_wmma_f32_16x16x4_f32(false, a, false, bm,
                                                  (short)0, d, false, false);
    }
    // D layout: VGPR r, lanes 0-15 -> M=r; lanes 16-31 -> M=r+8 (all N columns identical)

    // ---- effect = sigmoid(dot + eb) ----
    float e[8];
#pragma unroll
    for (int r = 0; r < 8; ++r) e[r] = 1.f / (1.f + __expf(-(d[r] + ebv)));

    // write effect: lane 0 has rows 0..7, lane 16 has rows 8..15 (column N=0)
    if ((lane & 15) == 0) {
        const int rb = tileBase + (lane >> 4) * 8;
#pragma unroll
        for (int r = 0; r < 8; ++r) out_effect[rb + r] = e[r];
    }

    // broadcast effects so every lane knows all 16 row effects
    float eo[8];
#pragma unroll
    for (int r = 0; r < 8; ++r) eo[r] = __shfl_xor(e[r], 16, 32);
    const bool hi = (lane >= 16);
    float eLow[8], eHigh[8];
#pragma unroll
    for (int r = 0; r < 8; ++r) {
        eLow[r]  = hi ? eo[r] : e[r];   // rows 0..7
        eHigh[r] = hi ? e[r] : eo[r];   // rows 8..15
    }

    // ---- out = effect * (tw * x_ + tb): single-use stream, non-temporal stores ----
#pragma unroll
    for (int r = 0; r < ROWS_PER_TILE; ++r) {
        const float er = (r < 8) ? eLow[r] : eHigh[r - 8];
        v4f o;
        o.x = er * fmaf(tw4.x, acc[r].x, tb4.x);
        o.y = er * fmaf(tw4.y, acc[r].y, tb4.y);
        o.z = er * fmaf(tw4.z, acc[r].z, tb4.z);
        o.w = er * fmaf(tw4.w, acc[r].w, tb4.w);
        __builtin_nontemporal_store(
            o, reinterpret_cast<v4f*>(out_y + (size_t)(tileBase + r) * DA_N + c0));
    }
}

extern "C" void kernel_launch(void* const* d_in, const int* in_sizes, int n_in,
                              void* d_out, int out_size, void* d_ws, size_t ws_size,
                              hipStream_t stream) {
    // inputs: 0=x (unused), 1=neigh_effect, 2=neigh_transform, 3=n_param,
    //         4=ew, 5=eb, 6=tw, 7=tb
    const float* ne = (const float*)d_in[1];
    const float* nt = (const float*)d_in[2];
    const float* np = (const float*)d_in[3];
    const float* ew = (const float*)d_in[4];
    const float* eb = (const float*)d_in[5];
    const float* tw = (const float*)d_in[6];
    const float* tb = (const float*)d_in[7];

    float* w_ws       = (float*)d_ws;          // [K,B] softmax weights (2 MB)
    float* out_effect = (float*)d_out;         // [B]
    float* out_y      = out_effect + B_N;      // [B,DA]

    softmax_kernel<<<B_N / 256, 256, 0, stream>>>(ne, w_ws);

    const int rowsPerBlock = ROWS_PER_TILE * TILES_PER_BLOCK;   // 64
    fused_kernel<<<B_N / rowsPerBlock, 32 * WAVES_PER_BLOCK, 0, stream>>>(
        nt, w_ws, np, ew, eb, tw, tb, out_effect, out_y);
}